// StrangeLoopLayer_50397146251291
// MI455X (gfx1250) — compile-verified
//
#include <hip/hip_runtime.h>
#include <hip/hip_bf16.h>

typedef __attribute__((ext_vector_type(2))) float v2f;
typedef __attribute__((ext_vector_type(8))) float v8f;

#define BATCH   500000
#define HIDDEN  64
#define FDIM    32
#define CHUNK   128
#define WARM    64
#define NTILES  (BATCH / 16)   // 31250, BATCH divisible by 16

__device__ __forceinline__ float fast_tanh(float x) {
    float xc = fminf(fmaxf(x, -9.0f), 9.0f);
    float e  = __expf(2.0f * xc);          // v_exp_f32 path
    return (e - 1.0f) / (e + 1.0f);
}

// ---------------------------------------------------------------------------
// Kernel 1: chunked strange-loop scan.
// Carry is (s1, s2) only (level 0 is overwritten each step). Contraction
// factor ~0.673/step => a 64-step warm-up from zero state reproduces the true
// carry to well below fp32 epsilon, letting 3907 threads run independently.
// ---------------------------------------------------------------------------
__global__ void strange_scan_kernel(const float* __restrict__ reality,
                                    const float* __restrict__ noise,
                                    float* __restrict__ ls /* [BATCH*3] */) {
    long c     = (long)blockIdx.x * blockDim.x + threadIdx.x;
    long start = c * CHUNK;
    if (start >= BATCH) return;
    long stop = start + CHUNK;
    if (stop > BATCH) stop = BATCH;
    long i = start - WARM;
    if (i < 0) i = 0;

    float s1 = 0.0f, s2 = 0.0f;
    for (; i < stop; ++i) {
        float r  = reality[i];
        float tr = fast_tanh(r);
        float ts = fast_tanh(s1);
        float n0 = noise[3 * i + 0];
        float n1 = noise[3 * i + 1];
        float n2 = noise[3 * i + 2];
        // new0 = 0.5*r + 0.2*(0.5*s1) + 0.05*n0
        float l0 = 0.5f * r  + 0.1f * s1 + 0.05f * n0;
        // new1 = 0.5*s1 + 0.3*tanh(r) + 0.2*(0.5*s2) + 0.05*n1
        float l1 = 0.5f * s1 + 0.3f * tr + 0.1f * s2 + 0.05f * n1;
        // new2 = 0.5*s2 + 0.3*tanh(s1) + 0.05*n2
        float l2 = 0.5f * s2 + 0.3f * ts + 0.05f * n2;
        s1 = l1;
        s2 = l2;
        if (i >= start) {
            ls[3 * i + 0] = l0;
            ls[3 * i + 1] = l1;
            ls[3 * i + 2] = l2;
        }
    }
}

// ---------------------------------------------------------------------------
// Kernel 2: MLP + LayerNorm with V_WMMA_F32_16X16X4_F32.
// One wave (32 lanes) owns one 16-row tile; 8 waves per block.
// All fragment loads are branchless (address = f(half)); the K=3 zero pad is
// a cndmask select, so EXEC stays all-ones with no saveexec churn.
// ---------------------------------------------------------------------------
__global__ __launch_bounds__(256)
void strange_mlp_ln_kernel(const float* __restrict__ ls,
                           const float* __restrict__ W1,
                           const float* __restrict__ b1,
                           const float* __restrict__ W2,
                           const float* __restrict__ b2,
                           const float* __restrict__ gamma,
                           const float* __restrict__ beta,
                           float* __restrict__ enc /* [BATCH*32] */) {
    __shared__ float hbuf[8][16][66];   // per-wave 16x64 h tile, padded stride

    const int wave = threadIdx.x >> 5;
    const int lane = threadIdx.x & 31;
    const int half = lane >> 4;         // 0: lanes 0-15, 1: lanes 16-31
    const int l    = lane & 15;

    const long tile   = (long)blockIdx.x * 8 + wave;   // wave-uniform
    const bool active = (tile < NTILES);               // wave-uniform branch
    const long m0     = tile * 16;

    v8f acc0 = {};
    v8f acc1 = {};

    if (active) {
        // ---- GEMM1: level_states [16x4] (K=3 padded) x W1 [4x64] ----
        // 32-bit A 16x4 layout: lanes 0-15 hold K0(v.x)/K1(v.y), lanes 16-31 K2/K3.
        const long row = m0 + l;
        v2f a;
        a.x = ls[row * 3 + 2 * half];                 // half0: K0, half1: K2
        {
            const float ay = ls[row * 3 + 1];         // K1 (unconditional load)
            a.y = half ? 0.0f : ay;                   // K3 pad = 0
        }

        #pragma unroll
        for (int t = 0; t < 4; ++t) {
            const int n0 = t * 16;
            v2f bm;   // B 4x16: row K striped across lanes (mirrors A layout)
            bm.x = W1[(2 * half) * HIDDEN + n0 + l];  // half0: K0, half1: K2
            {
                const float by = W1[1 * HIDDEN + n0 + l];
                bm.y = half ? 0.0f : by;              // K3 pad = 0
            }
            v8f cz = {};
            cz = __builtin_amdgcn_wmma_f32_16x16x4_f32(false, a, false, bm,
                                                       (short)0, cz, false, false);
            const float bias = b1[n0 + l];   // column = l for both halves
            #pragma unroll
            for (int j = 0; j < 8; ++j) {
                // C layout: element (M = j + 8*half, N = n0 + l)
                hbuf[wave][j + 8 * half][n0 + l] = fmaxf(cz[j] + bias, 0.0f);
            }
        }
    }
    __syncthreads();

    if (active) {
        // ---- GEMM2: h [16x64] x W2 [64x32], K blocked by 4 ----
        #pragma unroll 4
        for (int k0 = 0; k0 < HIDDEN; k0 += 4) {
            const int kk = k0 + 2 * half;        // A layout K split across halves
            v2f a;
            a.x = hbuf[wave][l][kk];
            a.y = hbuf[wave][l][kk + 1];
            v2f bA, bB;
            bA.x = W2[(kk)     * FDIM + l];       bA.y = W2[(kk + 1) * FDIM + l];
            bB.x = W2[(kk)     * FDIM + 16 + l];  bB.y = W2[(kk + 1) * FDIM + 16 + l];
            acc0 = __builtin_amdgcn_wmma_f32_16x16x4_f32(false, a, false, bA,
                                                         (short)0, acc0, false, false);
            acc1 = __builtin_amdgcn_wmma_f32_16x16x4_f32(false, a, false, bB,
                                                         (short)0, acc1, false, false);
        }

        const float bb0 = b2[l],        bb1 = b2[16 + l];
        const float g0  = gamma[l],     g1  = gamma[16 + l];
        const float be0 = beta[l],      be1 = beta[16 + l];

        #pragma unroll
        for (int j = 0; j < 8; ++j) {
            float e0 = acc0[j] + bb0;   // column l       of row (j + 8*half)
            float e1 = acc1[j] + bb1;   // column 16 + l  of same row
            float s = e0 + e1;
            float q = e0 * e0 + e1 * e1;
            // Row lives across a 16-lane half-wave: masks 1..8 stay inside it.
            #pragma unroll
            for (int m = 1; m < 16; m <<= 1) {
                s += __shfl_xor(s, m, 32);
                q += __shfl_xor(q, m, 32);
            }
            const float mu   = s * (1.0f / 32.0f);
            const float var  = q * (1.0f / 32.0f) - mu * mu;
            const float rstd = rsqrtf(var + 1e-5f);
            const long  row  = m0 + j + 8 * half;
            // Streaming output: non-temporal stores (TH=NT), never re-read here.
            __builtin_nontemporal_store((e0 - mu) * rstd * g0 + be0,
                                        &enc[row * FDIM + l]);
            __builtin_nontemporal_store((e1 - mu) * rstd * g1 + be1,
                                        &enc[row * FDIM + 16 + l]);
        }
    }
}

extern "C" void kernel_launch(void* const* d_in, const int* in_sizes, int n_in,
                              void* d_out, int out_size, void* d_ws, size_t ws_size,
                              hipStream_t stream) {
    (void)in_sizes; (void)n_in; (void)out_size; (void)d_ws; (void)ws_size;

    const float* reality = (const float*)d_in[0];
    const float* noise   = (const float*)d_in[1];
    const float* W1      = (const float*)d_in[2];
    const float* b1      = (const float*)d_in[3];
    const float* W2      = (const float*)d_in[4];
    const float* b2      = (const float*)d_in[5];
    const float* gamma   = (const float*)d_in[6];
    const float* beta    = (const float*)d_in[7];

    float* out = (float*)d_out;
    float* ls  = out;                           // [BATCH*3]
    float* enc = out + (size_t)BATCH * 3;       // [BATCH*32]

    const int nchunks = (BATCH + CHUNK - 1) / CHUNK;            // 3907
    strange_scan_kernel<<<(nchunks + 255) / 256, 256, 0, stream>>>(reality, noise, ls);

    const int nblocks = (NTILES + 7) / 8;                       // 3907
    strange_mlp_ln_kernel<<<nblocks, 256, 0, stream>>>(ls, W1, b1, W2, b2,
                                                       gamma, beta, enc);
}